// SurfaceSDF_86414741996105
// MI455X (gfx1250) — compile-verified
//
#include <hip/hip_runtime.h>
#include <hip/hip_bf16.h>

typedef __attribute__((ext_vector_type(2))) float v2f;
typedef __attribute__((ext_vector_type(8))) float v8f;

#define LOG2E 1.4426950408889634f
#define LN2   0.6931471805599453f
#define EPS   1e-12f

// One wave (32 lanes) owns 16 X-rows and sweeps all anchors in 16-wide tiles.
// Distance^2 comes out of a single V_WMMA_F32_16X16X4_F32 per tile:
//   A (16x4): [x0, x1, x2, |x|^2]   (K=0,1 in lanes 0-15; K=2,3 in lanes 16-31)
//   B (4x16): [-2a0, -2a1, -2a2, 1] (same K striping, N = lane&15)
//   C init  : |a_n|^2               (N fixed per lane -> splat across 8 VGPRs)
//   D       : ||x_m - a_n||^2       (M = vgpr + 8*(lane>=16), N = lane&15)
__global__ __launch_bounds__(256) void surface_sdf_kernel(
    const float* __restrict__ X, const float* __restrict__ A,
    const float* __restrict__ SIG, float* __restrict__ out,
    int N, int Na)
{
    __shared__ float4 sB[2048];   // {-2ax, -2ay, -2az, |a|^2}
    __shared__ float2 sS[2048];   // {sigma, log2e/(sigma+eps)}

    const int tid = threadIdx.x;
    for (int j = tid; j < Na; j += 256) {
        float ax = A[3*j+0], ay = A[3*j+1], az = A[3*j+2];
        sB[j] = make_float4(-2.0f*ax, -2.0f*ay, -2.0f*az, ax*ax + ay*ay + az*az);
        float sg = SIG[j];
        sS[j] = make_float2(sg, LOG2E / (sg + EPS));
    }
    __syncthreads();

    const int lane   = tid & 31;
    const int warp   = tid >> 5;
    const bool hi    = lane >= 16;          // lanes 16-31 carry K=2,3 / M+8
    const int  l16   = lane & 15;
    const int rowBase = blockIdx.x * 128 + warp * 16;
    const int m      = rowBase + l16;       // both halves load the same row

    const float xx = X[3*m+0], xy = X[3*m+1], xz = X[3*m+2];
    const float x2 = xx*xx + xy*xy + xz*xz;

    v2f amat;                                // A-matrix: 2 VGPRs
    amat.x = hi ? xz : xx;                   // K=2 : K=0
    amat.y = hi ? x2 : xy;                   // K=3 : K=1

    float num[8], den[8], s2[8];
    #pragma unroll
    for (int i = 0; i < 8; ++i) { num[i] = 0.f; den[i] = 0.f; s2[i] = 0.f; }

    const int ntiles = Na >> 4;              // 128 tiles of 16 anchors
    for (int nt = 0; nt < ntiles; ++nt) {
        const int n  = (nt << 4) + l16;
        const float4 b4 = sB[n];
        const float2 ss = sS[n];

        v2f bmat;                            // B-matrix: 2 VGPRs
        bmat.x = hi ? b4.z : b4.x;           // K=2 : K=0
        bmat.y = hi ? 1.0f : b4.y;           // K=3 : K=1

        v8f c;                               // C = |a_n|^2 (splat: N fixed per lane)
        #pragma unroll
        for (int i = 0; i < 8; ++i) c[i] = b4.w;

        c = __builtin_amdgcn_wmma_f32_16x16x4_f32(
                /*neg_a=*/false, amat, /*neg_b=*/false, bmat,
                /*c_mod=*/(short)0, c, /*reuse_a=*/false, /*reuse_b=*/false);

        const float sg = ss.x;               // sigma_n
        const float rr = ss.y;               // log2e/(sigma_n+eps)
        #pragma unroll
        for (int i = 0; i < 8; ++i) {
            float d2 = c[i] > 0.f ? c[i] : 0.f;
            float d  = __builtin_amdgcn_sqrtf(d2);
            float e1 = __builtin_amdgcn_exp2f(-d * LOG2E);   // exp(-d)
            num[i] += e1 * sg;
            den[i] += e1;
            s2[i]  += __builtin_amdgcn_exp2f(-d * rr);       // exp(-d/(sig+eps))
        }
    }

    // Reduce over N: each half's 16 lanes hold N=0..15 for the same M set.
    // xor masks 1,2,4,8 keep the shuffle inside each 16-lane half on wave32.
    #pragma unroll
    for (int mask = 1; mask < 16; mask <<= 1) {
        #pragma unroll
        for (int i = 0; i < 8; ++i) {
            num[i] += __shfl_xor(num[i], mask, 32);
            den[i] += __shfl_xor(den[i], mask, 32);
            s2[i]  += __shfl_xor(s2[i],  mask, 32);
        }
    }

    if (l16 == 0) {                          // lane 0 -> M=0..7, lane 16 -> M=8..15
        const int r0 = rowBase + (hi ? 8 : 0);
        #pragma unroll
        for (int i = 0; i < 8; ++i) {
            float f   = num[i] / (den[i] + EPS);
            float lse = __builtin_amdgcn_logf(s2[i]) * LN2;  // ln(sum exp)
            out[r0 + i] = -f * lse;
        }
    }
}

extern "C" void kernel_launch(void* const* d_in, const int* in_sizes, int n_in,
                              void* d_out, int out_size, void* d_ws, size_t ws_size,
                              hipStream_t stream) {
    const float* X   = (const float*)d_in[0];
    const float* A   = (const float*)d_in[1];
    const float* SIG = (const float*)d_in[2];
    float* out = (float*)d_out;

    const int N  = in_sizes[0] / 3;          // 65536
    const int Na = (n_in > 2) ? in_sizes[2] : 2048;  // 2048 (fits 48KB LDS staging)

    const int blocks = N / 128;              // 8 waves/block * 16 rows/wave
    surface_sdf_kernel<<<blocks, 256, 0, stream>>>(X, A, SIG, out, N, Na);
}